// Model_23880018165863
// MI455X (gfx1250) — compile-verified
//
#include <hip/hip_runtime.h>
#include <hip/hip_bf16.h>
#include <cstdint>
#include <math.h>

// Segment-max over uniform-degree CSR graph:
//   out[n, :] = max over neighbors c in col_idx[row_ptr[n]:row_ptr[n+1]] of node_feat[c, :]
// FEAT_DIM = 64 floats (256 B/row). Fast path (deg==16) uses CDNA5 async
// global->LDS copies (GLOBAL_LOAD_ASYNC_TO_LDS_B128, ASYNCcnt) to stage the
// 16 gathered rows (4 KB) per wave into LDS, then reduces from LDS.
// Wave-uniform control (node id, row_ptr, col_idx) is forced onto the scalar
// path via readfirstlane so row_ptr/col_idx come in through SMEM and the
// degree check is a scalar branch (async ops then execute with full EXEC).

#define WAVES_PER_BLOCK 8
#define BLOCK_THREADS   (WAVES_PER_BLOCK * 32)
#define FEAT            64      // floats per row
#define ROW_BYTES       256     // FEAT * 4
#define DEG             16
#define WAVE_BUF_FLOATS (DEG * FEAT)   // 1024 floats = 4 KB per wave

__global__ __launch_bounds__(BLOCK_THREADS)
void Model_23880018165863_segmax(const int* __restrict__ row_ptr,
                                 const int* __restrict__ col_idx,
                                 const float* __restrict__ feat,
                                 float* __restrict__ out,
                                 int n_nodes) {
  __shared__ float lds[WAVES_PER_BLOCK * WAVE_BUF_FLOATS];  // 32 KB

  const int lane = (int)(threadIdx.x & 31u);
  const int wave = (int)(threadIdx.x >> 5u);
  int node = (int)blockIdx.x * WAVES_PER_BLOCK + wave;
  if (node >= n_nodes) return;           // wave-uniform exit: EXEC stays full
  node = __builtin_amdgcn_readfirstlane(node);  // tell the compiler it's uniform

  const int s   = row_ptr[node];         // SMEM: uniform address
  const int e   = row_ptr[node + 1];
  const int deg = e - s;

  float2 acc;
  acc.x = -INFINITY;
  acc.y = -INFINITY;

  float* buf = lds + wave * WAVE_BUF_FLOATS;

  if (deg == DEG) {                      // scalar compare + branch
    // ---- async gather: 16 rows x 256 B -> LDS, 8 x b128 per wave ----
    // Generic pointer to __shared__ carries the LDS byte offset in its low
    // 32 bits (flat-aperture encoding); that is the VDST operand of the
    // async-to-LDS instruction (HW adds LDS_BASE).
    const uint32_t lds_base = (uint32_t)(uintptr_t)buf;
    const uint64_t gbase    = (uint64_t)(uintptr_t)feat;
    const int      half     = lane >> 4;   // 0: lanes 0-15, 1: lanes 16-31
    const int      chunk    = lane & 15;   // 16-byte chunk within a row

#pragma unroll
    for (int i = 0; i < 8; ++i) {
      // Both neighbor columns are wave-uniform -> scalar loads.
      const int colEven = col_idx[s + 2 * i];
      const int colOdd  = col_idx[s + 2 * i + 1];
      const int col     = half ? colOdd : colEven;        // v_cndmask
      const uint64_t gsrc = gbase + (uint64_t)(uint32_t)col * ROW_BYTES
                                  + (uint32_t)(chunk * 16);
      const uint32_t ldst = lds_base
                          + (uint32_t)((i * 32 + lane) * 16); // chunk c*16 bytes
      asm volatile("global_load_async_to_lds_b128 %0, %1, off"
                   :
                   : "v"(ldst), "v"(gsrc)
                   : "memory");
    }

    // Wait for all async copies of this wave to land in LDS.
    asm volatile("s_wait_asynccnt 0" ::: "memory");

    // ---- reduce: lane owns dims [2*lane, 2*lane+1] ----
    const float2* bufv = (const float2*)buf;   // row j = elements [j*32, j*32+32)
#pragma unroll
    for (int j = 0; j < DEG; ++j) {
      const float2 v = bufv[j * 32 + lane];    // ds_load, conflict-free
      acc.x = fmaxf(acc.x, v.x);
      acc.y = fmaxf(acc.y, v.y);
    }
  } else {
    // Generic-degree fallback (not taken for this input): direct gather.
    for (int j = 0; j < deg; ++j) {
      const int     col = col_idx[s + j];      // uniform -> scalar load
      const float2  v   = ((const float2*)(feat + (size_t)col * FEAT))[lane];
      acc.x = fmaxf(acc.x, v.x);
      acc.y = fmaxf(acc.y, v.y);
    }
  }

  ((float2*)out)[(size_t)node * (FEAT / 2) + lane] = acc;
}

extern "C" void kernel_launch(void* const* d_in, const int* in_sizes, int n_in,
                              void* d_out, int out_size, void* d_ws, size_t ws_size,
                              hipStream_t stream) {
  (void)n_in; (void)out_size; (void)d_ws; (void)ws_size;

  const int*   row_ptr = (const int*)d_in[0];    // [N_NODES + 1]
  const int*   col_idx = (const int*)d_in[1];    // [N_EDGES]
  const float* feat    = (const float*)d_in[2];  // [N_NODES, 64]
  float*       out     = (float*)d_out;          // [N_NODES, 64]

  const int n_nodes = in_sizes[0] - 1;
  const int grid    = (n_nodes + WAVES_PER_BLOCK - 1) / WAVES_PER_BLOCK;

  Model_23880018165863_segmax<<<grid, BLOCK_THREADS, 0, stream>>>(
      row_ptr, col_idx, feat, out, n_nodes);
}